// Hermite_12730283065702
// MI455X (gfx1250) — compile-verified
//
#include <hip/hip_runtime.h>
#include <cstddef>

#define NB 64        // N_BASIS
#define NF 16        // N_FUNCTIONS
#define NGRID 2097152

typedef __attribute__((ext_vector_type(2))) float v2f;
typedef __attribute__((ext_vector_type(8))) float v8f;

// compile-time sqrt (Newton, double)
constexpr double csqrt_d(double x) {
  if (x <= 0.0) return 0.0;
  double g = x > 1.0 ? x : 1.0;
  for (int i = 0; i < 64; ++i) g = 0.5 * (g + x / g);
  return g;
}

// Stable (orthonormal) Hermite-function recurrence coefficients:
//   q_{n+1} = a[n]*x*q_n - b[n]*q_{n-1},  a[n]=sqrt(2/(n+1)), b[n]=sqrt(n/(n+1))
struct Coefs {
  float a[2 * NB];
  float b[2 * NB];
  constexpr Coefs() : a{}, b{} {
    for (int n = 0; n < 2 * NB; ++n) {
      a[n] = (float)csqrt_d(2.0 / (double)(n + 1));
      b[n] = (float)csqrt_d((double)n / (double)(n + 1));
    }
  }
};
__constant__ Coefs COEFS{};

constexpr float SIGMA = 0.625f;  // R_MAX / sqrt(4*NB) = 10/16
constexpr float CPREF =
    (float)csqrt_d(0.625 * 2.0 * csqrt_d(3.14159265358979323846));  // sqrt(sigma*2*sqrt(pi))
constexpr float SQRT2 = (float)csqrt_d(2.0);

// One wave -> 16 grid points. Lanes 0-15 and 16-31 duplicate the recurrence for
// the same 16 points so B fragments for V_WMMA_F32_16X16X4_F32 are built with
// lane-conditional selects only.
__global__ __launch_bounds__(256) void hermite_wmma_kernel(
    const float* __restrict__ G, const float* __restrict__ W,
    float* __restrict__ out) {
  const int lane = threadIdx.x & 31;
  const int wave = threadIdx.x >> 5;
  const int half = lane >> 4;   // 0: lanes 0-15, 1: lanes 16-31
  const int p = lane & 15;      // grid point within tile / W row
  const int g0 = (blockIdx.x * 8 + wave) * 16;

  const float x = SIGMA * G[g0 + p];
  const float E = CPREF * expf(-0.5f * x * x);  // envelope * global prefactor

  // Preload A fragments from W (16 x 64, row-major).
  // ISA A-layout (32-bit, 16x4): lanes<16: {K=0,K=1}; lanes>=16: {K=2,K=3}
  const float* wrow = W + p * NB + 2 * half;
  v2f A[16];
#pragma unroll
  for (int s = 0; s < 16; ++s)
    A[s] = *(const v2f*)(wrow + 4 * s);  // global_load_b64

  // q'_n = q_n * envelope * prefactor (folded into the initial conditions)
  float qm = E;               // q'_0
  float qn = SQRT2 * x * E;   // q'_1
  v8f acc = {0.f, 0.f, 0.f, 0.f, 0.f, 0.f, 0.f, 0.f};

#pragma unroll
  for (int s = 0; s < 16; ++s) {
    // Slice s covers basis b = 4s..4s+3  (even orders 8s, 8s+2, 8s+4, 8s+6)
    float e0 = qm, e1 = 0.f, e2 = 0.f, e3 = 0.f;  // q'_{8s}, +2, +4, +6
    const int jmax = (s == 15) ? 5 : 8;           // stop at order 126
#pragma unroll
    for (int j = 0; j < jmax; ++j) {
      const int n = 8 * s + 1 + j;
      const float q2 = COEFS.a[n] * x * qn - COEFS.b[n] * qm;
      qm = qn;
      qn = q2;                 // qn = q'_{8s+2+j}
      if (j == 0) e1 = qn;
      if (j == 2) e2 = qn;
      if (j == 4) e3 = qn;
    }
    // B (4x16): VGPR0 = K-rows 0|2, VGPR1 = K-rows 1|3 across the half-waves
    v2f B;
    B.x = half ? e2 : e0;
    B.y = half ? e3 : e1;
    // D(16x16) += A(16x4) x B(4x16), f32 matrix pipe
    acc = __builtin_amdgcn_wmma_f32_16x16x4_f32(false, A[s], false, B,
                                                (short)0, acc, false, false);
  }

  // D layout: VGPR j -> function row j (lanes 0-15) / j+8 (lanes 16-31), N = p
  float* ob = out + (size_t)g0 + p + (size_t)(half * 8) * NGRID;
#pragma unroll
  for (int j = 0; j < 8; ++j)
    ob[(size_t)j * NGRID] = acc[j];
}

extern "C" void kernel_launch(void* const* d_in, const int* /*in_sizes*/, int /*n_in*/,
                              void* d_out, int /*out_size*/, void* /*d_ws*/, size_t /*ws_size*/,
                              hipStream_t stream) {
  const float* G = (const float*)d_in[0];   // (NGRID,) f32
  const float* W = (const float*)d_in[1];   // (NF, NB) f32
  float* out = (float*)d_out;               // (NF, NGRID) f32
  const int blocks = NGRID / (16 * 8);      // 16 pts per wave, 8 waves per block
  hermite_wmma_kernel<<<blocks, 256, 0, stream>>>(G, W, out);
}